// GRU_Aggregator_41918880809186
// MI455X (gfx1250) — compile-verified
//
#include <hip/hip_runtime.h>

typedef float v2f __attribute__((ext_vector_type(2)));
typedef float v4f __attribute__((ext_vector_type(4)));
typedef float v8f __attribute__((ext_vector_type(8)));

// Problem constants: B=16, N=64, D=32, F=32
#define PITCH 34   // LDS row pitch (floats): odd*2 -> conflict-free 16-lane column reads, 8B aligned

__global__ __launch_bounds__(256) void ecc_aggregate_kernel(
    const float* __restrict__ nodes,   // [16, 64, 32]
    const float* __restrict__ edges,   // [16, 4096, 32]
    const float* __restrict__ mask,    // [16, 4096]
    const float* __restrict__ W,       // [1024, 32]
    const float* __restrict__ bias,    // [1024]
    float* __restrict__ out)           // [16, 64, 32]
{
    __shared__ float sE[64 * PITCH];   // edge features for this (b,i), [j][f]
    __shared__ float sN[64 * PITCH];   // node features for this b,     [j][e]
    __shared__ float sMask[64];
    __shared__ float sPart[64];        // one partial per g-tile

    const int tid  = threadIdx.x;
    const int wave = tid >> 5;
    const int lane = tid & 31;
    const int lg   = lane >> 4;        // lane half (K/row-half select)
    const int m    = lane & 15;

    const int wg = blockIdx.x;         // b*64 + i
    const int b  = wg >> 6;
    const int i  = wg & 63;

    const float* eBase = edges + ((size_t)b * 4096 + (size_t)i * 64) * 32; // 64 contiguous edge rows
    const float* nBase = nodes + (size_t)b * 64 * 32;
    const float* mBase = mask  + (size_t)b * 4096 + (size_t)i * 64;

    // Cooperative stage into LDS (both blocks are fully contiguous in global)
    for (int q = tid; q < 64 * 32; q += 256) {
        const int j = q >> 5, f = q & 31;
        sE[j * PITCH + f] = eBase[q];
        sN[j * PITCH + f] = nBase[q];
    }
    if (tid < 64) sMask[tid] = mBase[tid];
    __syncthreads();

    float tileAcc[8];
#pragma unroll
    for (int k = 0; k < 8; ++k) tileAcc[k] = 0.0f;

#pragma unroll
    for (int k = 0; k < 8; ++k) {
        const int t  = wave + 8 * k;   // g-tile index 0..63 ; rows g = 16t .. 16t+15
        const int g0 = t << 4;
        const int e0 = (t & 1) << 4;   // e-half this tile covers (g = 32*d' + e)

        // A fragments: W[g0+m][4*kk + 2*lg + {0,1}] per the f32 16x4 layout
        v2f af[8];
#pragma unroll
        for (int kk = 0; kk < 8; ++kk)
            af[kk] = *(const v2f*)(W + (size_t)(g0 + m) * 32 + 4 * kk + 2 * lg);

        // Bias for the C rows this lane will hold: g0 + 8*lg + r, r=0..7
        const v4f b0 = *(const v4f*)(bias + g0 + 8 * lg);
        const v4f b1 = *(const v4f*)(bias + g0 + 8 * lg + 4);
        const float bv[8] = {b0.x, b0.y, b0.z, b0.w, b1.x, b1.y, b1.z, b1.w};

#pragma unroll
        for (int jg = 0; jg < 4; ++jg) {
            const int j16 = jg * 16 + m;        // this lane's column (edge j)
            v8f c = {};
#pragma unroll
            for (int kk = 0; kk < 8; ++kk) {
                // B fragment: E^T[K = 4*kk + 2*lg + {0,1}][N = j16]
                v2f bf = *(const v2f*)&sE[j16 * PITCH + 4 * kk + 2 * lg];
                c = __builtin_amdgcn_wmma_f32_16x16x4_f32(
                        false, af[kk], false, bf, (short)0, c, false, false);
            }
            // Consume C: row (vgpr r) = g0 + r + 8*lg = 32*d' + (e0 + r + 8*lg)
            float partial = 0.0f;
#pragma unroll
            for (int r = 0; r < 8; ++r) {
                float v = c[r] + bv[r];
                v = v > 0.0f ? v : 0.0f;                       // ReLU
                partial += v * sN[j16 * PITCH + e0 + 8 * lg + r];
            }
            partial *= sMask[j16];
            // Butterfly: sum over 16 j's and both e-subranges held across the wave
#pragma unroll
            for (int off = 16; off > 0; off >>= 1)
                partial += __shfl_xor(partial, off, 32);
            tileAcc[k] += partial;                             // accumulate over j-groups
        }
    }

    if (lane == 0) {
#pragma unroll
        for (int k = 0; k < 8; ++k) sPart[wave + 8 * k] = tileAcc[k];
    }
    __syncthreads();

    // out[b,i,d'] = tile(2d') + tile(2d'+1)  (the two e-halves)
    if (tid < 32)
        out[(size_t)wg * 32 + tid] = sPart[2 * tid] + sPart[2 * tid + 1];
}

extern "C" void kernel_launch(void* const* d_in, const int* in_sizes, int n_in,
                              void* d_out, int out_size, void* d_ws, size_t ws_size,
                              hipStream_t stream) {
    const float* nodes = (const float*)d_in[0];
    const float* edges = (const float*)d_in[1];
    const float* mask  = (const float*)d_in[2];
    const float* W     = (const float*)d_in[3];
    const float* bias  = (const float*)d_in[4];
    float* out         = (float*)d_out;

    ecc_aggregate_kernel<<<dim3(16 * 64), dim3(256), 0, stream>>>(
        nodes, edges, mask, W, bias, out);
}